// BahdanauAttnDecoderRNN_33749853012676
// MI455X (gfx1250) — compile-verified
//
#include <hip/hip_runtime.h>
#include <hip/hip_bf16.h>
#include <math.h>

#define H 512
#define V 50257
#define B 64
#define S 2048
#define L 2
#define CH 16          // context partial-sum chunks
#define SC (S / CH)

#define NT 4           // n-tiles per wave in WMMA GEMM
#define GW 8           // waves per GEMM block
#define MAXK (2 * H)   // largest K used by any GEMM call
#define LDAPAD 4       // LDS row padding (floats) -> conflict-free ds loads

typedef __attribute__((ext_vector_type(2))) float v2f;
typedef __attribute__((ext_vector_type(4))) float v4f;
typedef __attribute__((ext_vector_type(8))) float v8f;

__device__ __forceinline__ float sigmoidf_(float x) { return 1.0f / (1.0f + expf(-x)); }

// ---------------- embedding gather: emb[b,h] = table[idx[b], h] ----------------
__global__ void k_embed(const int* __restrict__ idx, const float* __restrict__ table,
                        float* __restrict__ emb) {
  int b = blockIdx.x, h = threadIdx.x;
  emb[b * H + h] = table[(size_t)idx[b] * H + h];
}

// ---------------- q[b,k] = sum_j h_top[b,j] * attn_W[j,k] ----------------
__global__ void k_query(const float* __restrict__ h_top, const float* __restrict__ attn_W,
                        float* __restrict__ q) {
  int b = blockIdx.x, k = threadIdx.x;
  float s = 0.f;
  for (int j = 0; j < H; ++j) s += h_top[b * H + j] * attn_W[j * H + k];
  q[b * H + k] = s;
}

// ---------------- qc[b] = h_top[b] . attn_b ----------------
__global__ void k_qbias(const float* __restrict__ h_top, const float* __restrict__ attn_b,
                        float* __restrict__ qc) {
  int b = threadIdx.x;
  float s = 0.f;
  for (int j = 0; j < H; ++j) s += h_top[b * H + j] * attn_b[j];
  qc[b] = s;
}

// ---------------- energies[b,s] = qc[b] + q[b,:] . enc[s,b,:] (one wave per pair) ----
// enc is streamed once (256 MB): non-temporal so it doesn't evict q / L2-resident data.
__global__ void k_energy(const float* __restrict__ enc, const float* __restrict__ q,
                         const float* __restrict__ qc, float* __restrict__ energy) {
  int wave = threadIdx.x >> 5;
  int lane = threadIdx.x & 31;
  int p = blockIdx.x * 8 + wave;          // p in [0, B*S)
  int b = p % B;
  int s = p / B;
  const v4f* e4 = (const v4f*)(enc + ((size_t)s * B + b) * H);
  const v4f* q4 = (const v4f*)(q + (size_t)b * H);
  float sum = 0.f;
#pragma unroll
  for (int i = 0; i < 4; ++i) {
    v4f e = __builtin_nontemporal_load(&e4[lane + 32 * i]);
    v4f qq = q4[lane + 32 * i];
    sum += e.x * qq.x + e.y * qq.y + e.z * qq.z + e.w * qq.w;
  }
#pragma unroll
  for (int off = 16; off > 0; off >>= 1) sum += __shfl_xor(sum, off, 32);
  if (lane == 0) energy[(size_t)b * S + s] = sum + qc[b];
}

// ---------------- row softmax over S ----------------
__global__ void k_softmax(const float* __restrict__ energy, float* __restrict__ attn) {
  __shared__ float red[256];
  int b = blockIdx.x, t = threadIdx.x;
  const float* row = energy + (size_t)b * S;
  float m = -INFINITY;
  for (int s = t; s < S; s += 256) m = fmaxf(m, row[s]);
  red[t] = m; __syncthreads();
  for (int o = 128; o > 0; o >>= 1) { if (t < o) red[t] = fmaxf(red[t], red[t + o]); __syncthreads(); }
  m = red[0]; __syncthreads();
  float sum = 0.f;
  for (int s = t; s < S; s += 256) sum += expf(row[s] - m);
  red[t] = sum; __syncthreads();
  for (int o = 128; o > 0; o >>= 1) { if (t < o) red[t] += red[t + o]; __syncthreads(); }
  float inv = 1.f / red[0];
  for (int s = t; s < S; s += 256) attn[(size_t)b * S + s] = expf(row[s] - m) * inv;
}

// ---------------- context partials (deterministic two-stage) ----------------
__global__ void k_ctx_partial(const float* __restrict__ enc, const float* __restrict__ attn,
                              float* __restrict__ partial) {
  int chunk = blockIdx.x, b = blockIdx.y, h = threadIdx.x;
  float acc = 0.f;
  int s0 = chunk * SC;
  for (int s = s0; s < s0 + SC; ++s)
    acc += attn[(size_t)b * S + s] *
           __builtin_nontemporal_load(&enc[((size_t)s * B + b) * H + h]);
  partial[((size_t)b * CH + chunk) * H + h] = acc;
}

__global__ void k_ctx_reduce(const float* __restrict__ partial, float* __restrict__ ctx) {
  int b = blockIdx.x, h = threadIdx.x;
  float acc = 0.f;
  for (int c = 0; c < CH; ++c) acc += partial[((size_t)b * CH + c) * H + h];
  ctx[b * H + h] = acc;
}

// ---------------- concat [a | c] along feature dim ----------------
__global__ void k_concat(const float* __restrict__ a, const float* __restrict__ c,
                         float* __restrict__ out) {
  int b = blockIdx.x, j = threadIdx.x;    // 2H threads
  out[b * (2 * H) + j] = (j < H) ? a[b * H + j] : c[b * H + j - H];
}

// ---------------- WMMA f32 GEMM: C[M,N] = (acc? C:0) + A[M,K] @ W[N,K]^T + bias[N] ---
// 256-thread blocks (8 waves). The 16xK A-tile for this block's m-tile is staged in
// LDS once (row stride K+4 -> lanes hit distinct banks on ds_load_b64). Each wave owns
// NT=4 consecutive 16x16 n-tiles: one shared A fragment per k-step feeds 4 independent
// V_WMMA_F32_16X16X4_F32 accumulator chains (4-way ILP hides the WMMA RAW latency).
// Weights are streamed non-temporal (no reuse, 206 MB for the vocab projection).
__global__ __launch_bounds__(256) void k_wmma_gemm_atb(
    const float* __restrict__ A, const float* __restrict__ W,
    const float* __restrict__ bias, float* __restrict__ C,
    int N, int K, int ldc, int accumulate) {
  __shared__ float As[16 * (MAXK + LDAPAD)];
  const int LDA = K + LDAPAD;
  int mtile = blockIdx.y;
  int row = mtile * 16;

  // cooperative stage of the A tile (coalesced along K)
  for (int r = 0; r < 16; ++r)
    for (int c = threadIdx.x; c < K; c += 256)
      As[r * LDA + c] = A[(size_t)(row + r) * K + c];
  __syncthreads();

  int wave = threadIdx.x >> 5;
  int lane = threadIdx.x & 31;
  int hi = lane >> 4;          // 0: K=k..k+1 / M low half,  1: K=k+2..k+3 / M high half
  int l16 = lane & 15;
  int tile0 = blockIdx.x * (GW * NT) + wave * NT;
  int m0 = row + (hi << 3);

  v8f acc[NT] = {};
  if (accumulate) {
#pragma unroll
    for (int t = 0; t < NT; ++t) {
      int nb = (tile0 + t) * 16 + l16;
      if (nb < N) {
#pragma unroll
        for (int r = 0; r < 8; ++r) acc[t][r] = C[(size_t)(m0 + r) * ldc + nb];
      }
    }
  }

  const float* as = &As[l16 * LDA + hi * 2];  // this lane's A row / K-half
  const v2f* wp[NT];
#pragma unroll
  for (int t = 0; t < NT; ++t) {
    int nb = (tile0 + t) * 16 + l16;
    int nclamp = nb < N ? nb : (N - 1);
    wp[t] = (const v2f*)(W + (size_t)nclamp * K);
  }

  for (int k = 0; k < K; k += 4) {
    v2f a;
    a.x = as[k];
    a.y = as[k + 1];
#pragma unroll
    for (int t = 0; t < NT; ++t) {
      v2f bfrag = __builtin_nontemporal_load(&wp[t][(k >> 1) + hi]);
      acc[t] = __builtin_amdgcn_wmma_f32_16x16x4_f32(
          /*neg_a=*/false, a, /*neg_b=*/false, bfrag,
          /*c_mod=*/(short)0, acc[t], /*reuse_a=*/false, /*reuse_b=*/false);
    }
  }

#pragma unroll
  for (int t = 0; t < NT; ++t) {
    int nb = (tile0 + t) * 16 + l16;
    if (nb < N) {
      float bv = bias ? bias[nb] : 0.f;
#pragma unroll
      for (int r = 0; r < 8; ++r)
        C[(size_t)(m0 + r) * ldc + nb] = acc[t][r] + bv;
    }
  }
}

// ---------------- LSTM pointwise (PyTorch gate order i,f,g,o) ----------------
__global__ void k_lstm_pw(const float* __restrict__ gates, const float* __restrict__ c_prev,
                          float* __restrict__ h_ws, float* __restrict__ h_out,
                          float* __restrict__ c_out) {
  int b = blockIdx.x, h = threadIdx.x;
  const float* g = gates + (size_t)b * 4 * H;
  float i = sigmoidf_(g[h]);
  float f = sigmoidf_(g[H + h]);
  float gg = tanhf(g[2 * H + h]);
  float o = sigmoidf_(g[3 * H + h]);
  float c = f * c_prev[b * H + h] + i * gg;
  float hn = o * tanhf(c);
  h_ws[b * H + h] = hn;
  h_out[b * H + h] = hn;
  c_out[b * H + h] = c;
}

// ---------------- in-place log-softmax over V ----------------
__global__ void k_logsoftmax(float* __restrict__ logits) {
  __shared__ float red[256];
  int b = blockIdx.x, t = threadIdx.x;
  float* row = logits + (size_t)b * V;
  float m = -INFINITY;
  for (int j = t; j < V; j += 256) m = fmaxf(m, row[j]);
  red[t] = m; __syncthreads();
  for (int o = 128; o > 0; o >>= 1) { if (t < o) red[t] = fmaxf(red[t], red[t + o]); __syncthreads(); }
  m = red[0]; __syncthreads();
  float sum = 0.f;
  for (int j = t; j < V; j += 256) sum += expf(row[j] - m);
  red[t] = sum; __syncthreads();
  for (int o = 128; o > 0; o >>= 1) { if (t < o) red[t] += red[t + o]; __syncthreads(); }
  float lse = m + logf(red[0]);
  for (int j = t; j < V; j += 256) row[j] = row[j] - lse;
}

extern "C" void kernel_launch(void* const* d_in, const int* in_sizes, int n_in,
                              void* d_out, int out_size, void* d_ws, size_t ws_size,
                              hipStream_t stream) {
  const int*   word  = (const int*)d_in[0];
  const float* h0    = (const float*)d_in[1];   // [L,B,H]
  const float* c0    = (const float*)d_in[2];   // [L,B,H]
  const float* enc   = (const float*)d_in[3];   // [S,B,H]
  const float* table = (const float*)d_in[4];   // [V,H]
  const float* attnW = (const float*)d_in[5];   // [H,H]
  const float* attnb = (const float*)d_in[6];   // [H]
  const float* Wih0  = (const float*)d_in[7];   // [4H,2H]
  const float* Whh0  = (const float*)d_in[8];   // [4H,H]
  const float* bih0  = (const float*)d_in[9];
  const float* bhh0  = (const float*)d_in[10];
  const float* Wih1  = (const float*)d_in[11];  // [4H,H]
  const float* Whh1  = (const float*)d_in[12];  // [4H,H]
  const float* bih1  = (const float*)d_in[13];
  const float* bhh1  = (const float*)d_in[14];
  const float* outW  = (const float*)d_in[15];  // [V,2H]
  const float* outb  = (const float*)d_in[16];  // [V]

  float* out = (float*)d_out;
  float* out_logp = out;                        // [B,V]
  float* out_h    = out + (size_t)B * V;        // [L,B,H]
  float* out_c    = out_h + (size_t)L * B * H;  // [L,B,H]
  float* out_attn = out_c + (size_t)L * B * H;  // [B,S]

  float* ws      = (float*)d_ws;
  float* emb     = ws;                    // B*H
  float* q       = emb + B * H;           // B*H
  float* qc      = q + B * H;             // B (pad 64)
  float* energy  = qc + 64;               // B*S
  float* partial = energy + B * S;        // B*CH*H
  float* ctx     = partial + B * CH * H;  // B*H
  float* rnn_in  = ctx + B * H;           // B*2H
  float* gates   = rnn_in + B * 2 * H;    // B*4H
  float* hl0     = gates + B * 4 * H;     // B*H
  float* hl1     = hl0 + B * H;           // B*H
  float* feat    = hl1 + B * H;           // B*2H

  const float* h_top = h0 + (size_t)(L - 1) * B * H;  // top layer hidden

  // Attention (algebraically folded: energy = (attn_W^T h_top).e + h_top.attn_b)
  k_embed<<<B, H, 0, stream>>>(word, table, emb);
  k_query<<<B, H, 0, stream>>>(h_top, attnW, q);
  k_qbias<<<1, B, 0, stream>>>(h_top, attnb, qc);
  k_energy<<<(B * S) / 8, 256, 0, stream>>>(enc, q, qc, energy);
  k_softmax<<<B, 256, 0, stream>>>(energy, out_attn);
  k_ctx_partial<<<dim3(CH, B), H, 0, stream>>>(enc, out_attn, partial);
  k_ctx_reduce<<<B, H, 0, stream>>>(partial, ctx);

  const int tilesPerBlock = GW * NT;  // 32 n-tiles (512 columns) per block
  // LSTM layer 0: gates = rnn_in@Wih0^T + b_ih0 + h0[0]@Whh0^T + b_hh0
  k_concat<<<B, 2 * H, 0, stream>>>(emb, ctx, rnn_in);
  {
    int ntiles = (4 * H) / 16;  // 128
    dim3 g((ntiles + tilesPerBlock - 1) / tilesPerBlock, B / 16);
    k_wmma_gemm_atb<<<g, 256, 0, stream>>>(rnn_in, Wih0, bih0, gates, 4 * H, 2 * H, 4 * H, 0);
    k_wmma_gemm_atb<<<g, 256, 0, stream>>>(h0, Whh0, bhh0, gates, 4 * H, H, 4 * H, 1);
    k_lstm_pw<<<B, H, 0, stream>>>(gates, c0, hl0, out_h, out_c);

    // LSTM layer 1
    k_wmma_gemm_atb<<<g, 256, 0, stream>>>(hl0, Wih1, bih1, gates, 4 * H, H, 4 * H, 0);
    k_wmma_gemm_atb<<<g, 256, 0, stream>>>(h0 + (size_t)B * H, Whh1, bhh1, gates, 4 * H, H, 4 * H, 1);
    k_lstm_pw<<<B, H, 0, stream>>>(gates, c0 + (size_t)B * H, hl1,
                                   out_h + (size_t)B * H, out_c + (size_t)B * H);
  }

  // Output projection (32 FLOP/byte -> compute-bound in f32 -> WMMA) + log-softmax
  k_concat<<<B, 2 * H, 0, stream>>>(hl1, ctx, feat);
  {
    int ntiles = (V + 15) / 16;  // 3142
    dim3 g((ntiles + tilesPerBlock - 1) / tilesPerBlock, B / 16);
    k_wmma_gemm_atb<<<g, 256, 0, stream>>>(feat, outW, outb, out_logp, V, 2 * H, V, 0);
  }
  k_logsoftmax<<<B, 256, 0, stream>>>(out_logp);
}